// LiftBlock_47820165874218
// MI455X (gfx1250) — compile-verified
//
#include <hip/hip_runtime.h>
#include <math.h>

// Problem constants (from the reference)
#define CIN  8
#define COUTC 32
#define EPSV 1e-6f

typedef __attribute__((ext_vector_type(2))) float v2f;
typedef __attribute__((ext_vector_type(8))) float v8f;

// D = A(16x4 f32) * B(4x16 f32) + C(16x16 f32), exact fp32 WMMA on gfx1250.
__device__ __forceinline__ v8f wmma_f32_k4(v2f a, v2f b, v8f c) {
  return __builtin_amdgcn_wmma_f32_16x16x4_f32(
      /*neg_a=*/false, a, /*neg_b=*/false, b,
      /*c_mod=*/(short)0, c, /*reuse_a=*/false, /*reuse_b=*/false);
}

// ---------------------------------------------------------------------------
// Kernel 0: zero the aggregation buffer (d_ws is poisoned by the harness).
// ---------------------------------------------------------------------------
__global__ void zero_kernel(float* __restrict__ p, int n) {
  int i = blockIdx.x * blockDim.x + threadIdx.x;
  if (i < n) p[i] = 0.0f;
}

// ---------------------------------------------------------------------------
// Kernel 1: per-node precompute  T[n, r*32+o] = sum_i x[n,i] * W_rad[r,i,o]
// A tiny [N x 8] @ [8 x 64] GEMM done with V_WMMA_F32_16X16X4_F32.
// One wave handles 16 nodes; K=8 -> 2 chained WMMAs per 16-col tile.
// Wave id goes through readfirstlane so tile guards are SCALAR branches.
// ---------------------------------------------------------------------------
__global__ void __launch_bounds__(128) precompute_T_kernel(
    const float* __restrict__ x, const float* __restrict__ W_rad,
    float* __restrict__ T, int n_nodes) {
  const int lane = threadIdx.x & 31;
  const int wv   = __builtin_amdgcn_readfirstlane((int)(threadIdx.x >> 5));
  const int wid  = blockIdx.x * 4 + wv;       // SGPR
  const int tbase = wid * 16;                 // SGPR
  if (tbase >= n_nodes) return;               // scalar branch, EXEC untouched

  const int half = lane >> 4;                 // lane half selects K sub-pair
  const int l16  = lane & 15;
  const bool full = (tbase + 16 <= n_nodes);  // SGPR condition

  int nodeA = tbase + l16;                    // A layout: lane -> row (M)
  if (nodeA >= n_nodes) nodeA = n_nodes - 1;  // clamp (v_min), no branch

  // A chunks: chunk kk covers K = 4kk..4kk+3 ; lane holds K = 4kk+2*half, +1
  v2f a[2];
#pragma unroll
  for (int kk = 0; kk < 2; ++kk) {
    int kb = 4 * kk + 2 * half;
    a[kk].x = x[nodeA * CIN + kb];
    a[kk].y = x[nodeA * CIN + kb + 1];
  }

#pragma unroll
  for (int cb = 0; cb < 4; ++cb) {            // 4 col tiles of 16 -> 64 outputs
    int col = cb * 16 + l16;                  // col = r*32 + o
    int r = col >> 5, o = col & 31;
    v8f acc = {0.f, 0.f, 0.f, 0.f, 0.f, 0.f, 0.f, 0.f};
#pragma unroll
    for (int kk = 0; kk < 2; ++kk) {
      int i0 = 4 * kk + 2 * half;             // B layout mirrors A's K split
      v2f b;
      b.x = W_rad[(r * CIN + i0) * COUTC + o];
      b.y = W_rad[(r * CIN + i0 + 1) * COUTC + o];
      acc = wmma_f32_k4(a[kk], b, acc);
    }
    // D layout: VGPR t -> row M = t + 8*half ; lane&15 -> column
    if (full) {                               // scalar fast path
#pragma unroll
      for (int t = 0; t < 8; ++t)
        T[(size_t)(tbase + t + 8 * half) * 64 + col] = acc[t];
    } else {                                  // rare scalar-branched tail
#pragma unroll
      for (int t = 0; t < 8; ++t) {
        int node = tbase + t + 8 * half;
        if (node < n_nodes) T[(size_t)node * 64 + col] = acc[t];
      }
    }
  }
}

// ---------------------------------------------------------------------------
// Kernel 2: edge stage. ONE WAVE PER EDGE; lane = output channel o.
// The edge index is forced into an SGPR (readfirstlane) so src/dst/pcmp
// become scalar (SMEM) loads; T[src] is two coalesced 128B wave loads; the
// two atomics per lane hit consecutive dwords -> cacheline-granular L2
// atomics. T and agg both live in the 192MB L2, never HBM.
// ---------------------------------------------------------------------------
__global__ void __launch_bounds__(256) edge_scatter_kernel(
    const int* __restrict__ ei, const float* __restrict__ pcmp,
    const float* __restrict__ T, float* __restrict__ agg, int n_edges) {
  const int lane = threadIdx.x & 31;
  const int wave = __builtin_amdgcn_readfirstlane((int)(threadIdx.x >> 5));
  const long long e = (long long)blockIdx.x * 8 + wave;   // SGPR
  if (e >= n_edges) return;                  // scalar branch

  const int src = ei[e];                       // uniform -> scalar load
  const int dst = ei[(long long)n_edges + e];  // uniform -> scalar load
  const float p00 = pcmp[e * 4 + 0];           // [e, r=0, re]
  const float p01 = pcmp[e * 4 + 1];           // [e, r=0, im]
  const float p10 = pcmp[e * 4 + 2];           // [e, r=1, re]
  const float p11 = pcmp[e * 4 + 3];           // [e, r=1, im]

  const float t0 = T[(size_t)src * 64 + lane];       // T[src, r=0, o=lane]
  const float t1 = T[(size_t)src * 64 + 32 + lane];  // T[src, r=1, o=lane]

  float* dp = &agg[(size_t)dst * 64 + 2 * lane];
  atomicAdd(dp,     fmaf(t0, p00, t1 * p10));  // re
  atomicAdd(dp + 1, fmaf(t0, p01, t1 * p11));  // im
}

// ---------------------------------------------------------------------------
// Kernel 3: node postprocess. One wave handles 8 nodes = 16 (node,re/im) rows.
// Phase rotation + nonlin, then 3x (32x32 complex-linear via 8 chained K=4
// f32 WMMAs per 16x16 tile, 2 col tiles) + nonlin. C/D layout puts (re,im)
// of each node in adjacent VGPRs of one lane -> per-lane magnitude ReLU.
// Inter-layer C->A transpose goes through a wave-private padded LDS tile.
// All tile-level guards are SGPR conditions (scalar branches only).
// ---------------------------------------------------------------------------
__global__ void __launch_bounds__(128) node_post_kernel(
    const float* __restrict__ agg,
    const float* __restrict__ phase, const float* __restrict__ b_nl,
    const float* __restrict__ W1, const float* __restrict__ b1,
    const float* __restrict__ W2, const float* __restrict__ b2,
    const float* __restrict__ W3, const float* __restrict__ b3,
    float* __restrict__ out, int n_nodes) {
  __shared__ float Y[4][16][33];            // per-wave 16x32 tile, padded
  const int lane = threadIdx.x & 31;
  const int wv   = __builtin_amdgcn_readfirstlane((int)(threadIdx.x >> 5));
  const int wid  = blockIdx.x * 4 + wv;     // SGPR
  const int nb   = wid * 8;                 // SGPR: first node of this wave
  const bool active = (nb < n_nodes);       // SGPR condition
  const bool full   = (nb + 8 <= n_nodes);  // SGPR condition
  const int half = lane >> 4, l16 = lane & 15;
  float (*Yw)[33] = Y[wv];

  // Stage 0: load agg, phase-rotate, nonlin(b_nl), write row-major tile.
  // Tail reads are clamped (duplicated rows); tail *writes* are guarded later.
  if (active) {
    float ph = phase[lane];                 // o = lane (0..31)
    float cp = __cosf(ph), sp = __sinf(ph);
    float bb = b_nl[lane];
#pragma unroll
    for (int p = 0; p < 8; ++p) {
      int node = nb + p;
      if (node >= n_nodes) node = n_nodes - 1;
      float re = agg[(size_t)node * 64 + lane * 2];
      float im = agg[(size_t)node * 64 + lane * 2 + 1];
      float rr = re * cp - im * sp;
      float ii = re * sp + im * cp;
      float mag = __fsqrt_rn(rr * rr + ii * ii + EPSV);
      float sc  = fmaxf(mag + bb, 0.f) / mag;
      Yw[2 * p][lane]     = rr * sc;        // row = node_local*2 + comp
      Yw[2 * p + 1][lane] = ii * sc;
    }
  }
  __syncthreads();

#pragma unroll 1
  for (int layer = 0; layer < 3; ++layer) {
    // Uniform pointer selects (s_cselect), no indexed pointer array.
    const float* W = (layer == 0) ? W1 : (layer == 1) ? W2 : W3;
    const float* b = (layer == 0) ? b1 : (layer == 1) ? b2 : b3;
    v8f acc0 = {0.f, 0.f, 0.f, 0.f, 0.f, 0.f, 0.f, 0.f};
    v8f acc1 = {0.f, 0.f, 0.f, 0.f, 0.f, 0.f, 0.f, 0.f};
    if (active) {
#pragma unroll
      for (int kk = 0; kk < 8; ++kk) {      // K = 32 in chunks of 4
        int i0 = 4 * kk + 2 * half;
        v2f a;                              // A: row = l16, K = i0, i0+1
        a.x = Yw[l16][i0];
        a.y = Yw[l16][i0 + 1];
        v2f bA, bB;                         // B: out cols [0,16) and [16,32)
        bA.x = W[i0 * COUTC + l16];
        bA.y = W[(i0 + 1) * COUTC + l16];
        bB.x = W[i0 * COUTC + 16 + l16];
        bB.y = W[(i0 + 1) * COUTC + 16 + l16];
        acc0 = wmma_f32_k4(a, bA, acc0);
        acc1 = wmma_f32_k4(a, bB, acc1);
      }
      // Magnitude ReLU: VGPR pairs (2t,2t+1) are (re,im) of one node.
      float bv0 = b[l16];
      float bv1 = b[16 + l16];
#pragma unroll
      for (int t = 0; t < 4; ++t) {
        float re = acc0[2 * t], im = acc0[2 * t + 1];
        float mag = __fsqrt_rn(re * re + im * im + EPSV);
        float sc  = fmaxf(mag + bv0, 0.f) / mag;
        acc0[2 * t] = re * sc; acc0[2 * t + 1] = im * sc;
        re = acc1[2 * t]; im = acc1[2 * t + 1];
        mag = __fsqrt_rn(re * re + im * im + EPSV);
        sc  = fmaxf(mag + bv1, 0.f) / mag;
        acc1[2 * t] = re * sc; acc1[2 * t + 1] = im * sc;
      }
    }
    __syncthreads();                        // reads above vs writes below
    if (layer < 2) {
      if (active) {
#pragma unroll
        for (int t = 0; t < 8; ++t) {       // C layout -> row-major LDS
          int M = t + 8 * half;
          Yw[M][l16]      = acc0[t];
          Yw[M][16 + l16] = acc1[t];
        }
      }
      __syncthreads();
    } else if (active) {
      if (full) {                           // scalar fast path: no exec ops
#pragma unroll
        for (int t = 0; t < 8; ++t) {       // final store [N, 32, 2]
          int M = t + 8 * half;
          size_t base = (size_t)(nb + (M >> 1)) * 64 + (M & 1);
          out[base + l16 * 2]        = acc0[t];
          out[base + (16 + l16) * 2] = acc1[t];
        }
      } else {                              // rare scalar-branched tail
#pragma unroll
        for (int t = 0; t < 8; ++t) {
          int M = t + 8 * half;
          int node = nb + (M >> 1);
          if (node < n_nodes) {
            size_t base = (size_t)node * 64 + (M & 1);
            out[base + l16 * 2]        = acc0[t];
            out[base + (16 + l16) * 2] = acc1[t];
          }
        }
      }
    }
  }
}

// ---------------------------------------------------------------------------
extern "C" void kernel_launch(void* const* d_in, const int* in_sizes, int n_in,
                              void* d_out, int out_size, void* d_ws, size_t ws_size,
                              hipStream_t stream) {
  const float* x     = (const float*)d_in[0];
  const int*   ei    = (const int*)d_in[1];
  const float* pcmp  = (const float*)d_in[2];
  const float* W_rad = (const float*)d_in[3];
  const float* phase = (const float*)d_in[4];
  const float* b_nl  = (const float*)d_in[5];
  const float* W1 = (const float*)d_in[6];
  const float* b1 = (const float*)d_in[7];
  const float* W2 = (const float*)d_in[8];
  const float* b2 = (const float*)d_in[9];
  const float* W3 = (const float*)d_in[10];
  const float* b3 = (const float*)d_in[11];

  const int n_nodes = in_sizes[0] / CIN;
  const int n_edges = in_sizes[1] / 2;

  // Workspace: T[N,64] then agg[N,64] (25.6 MB total at N=50000).
  float* T   = (float*)d_ws;
  float* agg = T + (size_t)n_nodes * 64;

  const int aggN = n_nodes * 64;
  zero_kernel<<<(aggN + 255) / 256, 256, 0, stream>>>(agg, aggN);

  const int nwaves_T  = (n_nodes + 15) / 16;
  const int nblocks_T = (nwaves_T + 3) / 4;
  precompute_T_kernel<<<nblocks_T, 128, 0, stream>>>(x, W_rad, T, n_nodes);

  const int eblocks = (n_edges + 7) / 8;    // one wave per edge, 8 waves/block
  edge_scatter_kernel<<<eblocks, 256, 0, stream>>>(ei, pcmp, T, agg, n_edges);

  const int nwaves_P  = (n_nodes + 7) / 8;
  const int nblocks_P = (nwaves_P + 3) / 4;
  node_post_kernel<<<nblocks_P, 128, 0, stream>>>(
      agg, phase, b_nl, W1, b1, W2, b2, W3, b3, (float*)d_out, n_nodes);
}